// Q8Model_63831803953403
// MI455X (gfx1250) — compile-verified
//
#include <hip/hip_runtime.h>
#include <hip/hip_bf16.h>

#define HH 768
#define WW 768
#define NPIX (HH * WW)
#define E_CNT 4096

typedef __attribute__((ext_vector_type(2))) float v2f;
typedef __attribute__((ext_vector_type(8))) float v8f;

// ---------------------------------------------------------------------------
// Kernel 1: per-pixel Q8 shape-function eval + bi-quintic surface eval +
// masked squared residual, scattered into per-element (ssum, cnt) with
// fp32 global atomics. One thread per pixel. The gather of the 6x6 Q tile is
// the dominant cost -> issue global_prefetch as soon as the address is known,
// overlap polynomial setup with the in-flight gather, load the tile as 9x b128.
// ---------------------------------------------------------------------------
__global__ __launch_bounds__(256) void q8_pixel_kernel(
    const float* __restrict__ gcoords,          // [H,W,2]  (x,y)
    const float* __restrict__ lcoords,          // [H,W,2]  (xi,eta)
    const float* __restrict__ refImg,           // [H,W]
    const float* __restrict__ Qtab,             // [H,W,6,6]
    const float* __restrict__ nodes_u,          // [NN]
    const float* __restrict__ nodes_v,          // [NN]
    const int*   __restrict__ elements,         // [E,8]
    const int*   __restrict__ threaddiagram,    // [H,W]
    const unsigned char* __restrict__ validpts, // [H,W] (bool, 1 byte)
    float* __restrict__ ssum,                   // [E]
    float* __restrict__ cnt)                    // [E]
{
    int p = blockIdx.x * blockDim.x + threadIdx.x;
    if (p >= NPIX) return;

    int e = threaddiagram[p];

    // Connectivity row: 32B, 16B-aligned -> two b128 loads.
    const int4* conn4 = (const int4*)(elements + e * 8);
    int4 cA = conn4[0];
    int4 cB = conn4[1];

    const float2* lc2 = (const float2*)lcoords;
    float2 le  = lc2[p];
    float  xi  = le.x;
    float  eta = le.y;
    float xi2 = xi * xi, eta2 = eta * eta;

    float N[8];
    N[0] = -0.25f * (1.f - xi) * (1.f - eta) * (1.f + xi + eta);
    N[1] = -0.25f * (1.f + xi) * (1.f - eta) * (1.f - xi + eta);
    N[2] = -0.25f * (1.f + xi) * (1.f + eta) * (1.f - xi - eta);
    N[3] = -0.25f * (1.f - xi) * (1.f + eta) * (1.f + xi - eta);
    N[4] =  0.5f  * (1.f - xi2) * (1.f - eta);
    N[5] =  0.5f  * (1.f + xi)  * (1.f - eta2);
    N[6] =  0.5f  * (1.f - xi2) * (1.f + eta);
    N[7] =  0.5f  * (1.f - xi)  * (1.f - eta2);

    int cn[8] = { cA.x, cA.y, cA.z, cA.w, cB.x, cB.y, cB.z, cB.w };
    float u = 0.f, v = 0.f;
#pragma unroll
    for (int i = 0; i < 8; ++i) {
        int n = cn[i];
        u = fmaf(N[i], nodes_u[n], u);
        v = fmaf(N[i], nodes_v[n], v);
    }

    const float2* gc2 = (const float2*)gcoords;
    float2 g = gc2[p];
    float x = g.x + u;
    float y = g.y + v;
    float xf = floorf(x), yf = floorf(y);
    int xidx = (int)xf; xidx = xidx < 0 ? 0 : (xidx > WW - 1 ? WW - 1 : xidx);
    int yidx = (int)yf; yidx = yidx < 0 ? 0 : (yidx > HH - 1 ? HH - 1 : yidx);

    const float* q = Qtab + (size_t)(yidx * WW + xidx) * 36;
    __builtin_prefetch(q, 0, 0);  // gfx1250 global_prefetch_b8: start the gather now

    float xd = x - xf, yd = y - yf;
    float xp[6], yp[6];
    xp[0] = 1.f; yp[0] = 1.f;
#pragma unroll
    for (int k = 1; k < 6; ++k) { xp[k] = xp[k - 1] * xd; yp[k] = yp[k - 1] * yd; }

    // 144B contiguous, 16B-aligned (36 floats * 4B, offset multiple of 144)
    float qv[36];
    const float4* q4 = (const float4*)q;
#pragma unroll
    for (int k = 0; k < 9; ++k) {
        float4 t = q4[k];
        qv[4 * k + 0] = t.x; qv[4 * k + 1] = t.y;
        qv[4 * k + 2] = t.z; qv[4 * k + 3] = t.w;
    }

    float val = 0.f;
#pragma unroll
    for (int i = 0; i < 6; ++i) {
        float row = 0.f;
#pragma unroll
        for (int j = 0; j < 6; ++j) row = fmaf(qv[6 * i + j], xp[j], row);
        val = fmaf(yp[i], row, val);
    }

    float d  = refImg[p] - val;
    float r2 = d * d;

    if (validpts[p]) {
        atomicAdd(&ssum[e], r2);
        atomicAdd(&cnt[e], 1.0f);
    }
}

// ---------------------------------------------------------------------------
// Kernel 2: single-wave WMMA segment reduction.
// loss = sum_e ssum[e] / max(cnt[e], 1).
// (Workspace is zeroed every call, so cnt[e]==0 implies ssum[e]==0 and the
//  reference's where(cnt>0, ...) select is exactly 0/1 = 0 -> no branch, no
//  EXEC divergence anywhere in the wave, as WMMA requires.)
// With A = ones(16x4), D = A x B + C makes every row of D the column-sums of
// B, so each V_WMMA_F32_16X16X4_F32 folds 64 values into the accumulator
// (placement inside B is irrelevant for a pure sum). 4096/64 = 64 WMMAs,
// then sum the 16 distinct entries of row 0 (lanes 0-15, VGPR 0 of C).
// ---------------------------------------------------------------------------
__global__ __launch_bounds__(32) void q8_reduce_wmma(
    const float* __restrict__ ssum,
    const float* __restrict__ cnt,
    float* __restrict__ out)
{
    int lane = threadIdx.x;  // 0..31, wave32

    v2f a; a.x = 1.0f; a.y = 1.0f;  // A = ones (16x4 f32 -> 2 VGPRs/lane)
    v8f c = {};                      // C/D accumulator (16x16 f32)

    for (int i = 0; i < E_CNT / 64; ++i) {
        int base = i * 64 + lane;
        float s0 = ssum[base],      c0 = cnt[base];
        float s1 = ssum[base + 32], c1 = cnt[base + 32];
        v2f b;  // B = 4x16 f32 -> 2 VGPRs/lane; each of the 64 values appears once
        b.x = s0 / fmaxf(c0, 1.0f);   // exact 0 when c0==0 (then s0==0 too)
        b.y = s1 / fmaxf(c1, 1.0f);
        // 8 args: (neg_a, A, neg_b, B, c_mod, C, reuse_a, reuse_b)
        c = __builtin_amdgcn_wmma_f32_16x16x4_f32(
                false, a, false, b, (short)0, c, false, false);
    }

    // Row 0 of D lives in VGPR 0: lanes 0..15 hold D[0, 0..15] (lanes 16..31
    // hold the duplicate row 8). Sum the 16 distinct column-sums.
    __shared__ float tmp[32];
    tmp[lane] = c[0];
    __syncthreads();
    if (lane == 0) {
        float s = 0.f;
#pragma unroll
        for (int n = 0; n < 16; ++n) s += tmp[n];
        *out = s;
    }
}

// ---------------------------------------------------------------------------
extern "C" void kernel_launch(void* const* d_in, const int* in_sizes, int n_in,
                              void* d_out, int out_size, void* d_ws, size_t ws_size,
                              hipStream_t stream) {
    const float*         gcoords       = (const float*)d_in[0];         // (H,W,2)
    const float*         lcoords       = (const float*)d_in[1];         // (H,W,2)
    const float*         refImg        = (const float*)d_in[2];         // (H,W)
    const float*         Qtab          = (const float*)d_in[3];         // (H,W,6,6)
    const float*         nodes_u       = (const float*)d_in[4];         // (NN,)
    const float*         nodes_v       = (const float*)d_in[5];         // (NN,)
    const int*           elements      = (const int*)d_in[6];           // (E,8)
    const int*           threaddiagram = (const int*)d_in[7];           // (H,W)
    const unsigned char* validpts      = (const unsigned char*)d_in[8]; // (H,W) bool

    float* ssum = (float*)d_ws;
    float* cnt  = ssum + E_CNT;

    // Zero the per-element accumulators every call (harness does not re-poison).
    hipMemsetAsync(d_ws, 0, 2 * E_CNT * sizeof(float), stream);

    q8_pixel_kernel<<<(NPIX + 255) / 256, 256, 0, stream>>>(
        gcoords, lcoords, refImg, Qtab, nodes_u, nodes_v,
        elements, threaddiagram, validpts, ssum, cnt);

    q8_reduce_wmma<<<1, 32, 0, stream>>>(ssum, cnt, (float*)d_out);
}